// SARRasterizer_21809843929259
// MI455X (gfx1250) — compile-verified
//
#include <hip/hip_runtime.h>
#include <math.h>

#define N_GAUSS 2048
#define RS 128
#define AS 128
#define NPIX (RS * AS)
#define CHUNK 512              // gaussians staged in LDS per pass (20KB)

typedef float v2f __attribute__((ext_vector_type(2)));
typedef float v8f __attribute__((ext_vector_type(8)));

// ---- workspace layout (float offsets) ----
#define WS_FEAT 0                       // 2048 * 8
#define WS_OP   (N_GAUSS * 8)           // 2048
#define WS_INT  (WS_OP + N_GAUSS)       // 2048
#define WS_KEY  (WS_INT + N_GAUSS)      // 2048
#define WS_ORD  (WS_KEY + N_GAUSS)      // 2048 ints

// ------------------------------------------------------------------
// Kernel 1: per-gaussian preprocessing -> rank-6 bilinear features
// ------------------------------------------------------------------
__global__ void sar_preprocess(const float* __restrict__ means,
                               const float* __restrict__ cov,
                               const float* __restrict__ trans,
                               const float* __restrict__ sh,
                               float* __restrict__ feat,
                               float* __restrict__ op,
                               float* __restrict__ inten,
                               float* __restrict__ rkey) {
    int n = blockIdx.x * blockDim.x + threadIdx.x;
    if (n >= N_GAUSS) return;

    // radar geometry (incidence 30deg, azimuth 0, track 0)
    const float p_rx = -5773.5026919f;   // -10000*tan(30deg)
    const float p_rz = 10000.0f;
    const float r0   = 11527.8053851f;   // |p_r| - 0.5*128*0.3
    const float RR = 0.3f, AR = 0.3f, BLUR = 0.3f;

    float mx = means[n * 3 + 0], my = means[n * 3 + 1], mz = means[n * 3 + 2];
    float dx = mx - p_rx, dy = my, dz = mz - p_rz;
    float r = sqrtf(dx * dx + dy * dy + dz * dz);
    float ir = 1.0f / r;
    float lx = dx * ir, ly = dy * ir, lz = dz * ir;

    float u = (r - r0) / RR;
    float v = mx / AR + 0.5f * (float)AS;   // track t = (1,0,0)

    // cov2 = J S J^T,  J = [los/RR ; t/AR]
    float jx = lx / RR, jy = ly / RR, jz = lz / RR;
    float ax = 1.0f / AR;
    const float* S = cov + n * 9;
    float s00 = S[0], s01 = S[1], s02 = S[2], s11 = S[4], s12 = S[5], s22 = S[8];
    float t0x = s00 * jx + s01 * jy + s02 * jz;
    float t0y = s01 * jx + s11 * jy + s12 * jz;
    float t0z = s02 * jx + s12 * jy + s22 * jz;
    float c00 = jx * t0x + jy * t0y + jz * t0z;
    float c01 = ax * t0x;
    float c11 = ax * ax * s00;

    float A = c00 + BLUR, B = c01, C = c11 + BLUR;
    float det = fmaxf(A * C - B * B, 1e-6f);
    float ca = C / det, cb = -B / det, cc = A / det;

    // rank-6 features: power = f0*pu^2 + f1*pu + f2*pv^2 + f3*pv + f4*pu*pv + f5
    float* f = feat + n * 8;
    f[0] = -0.5f * ca;
    f[1] = ca * u + cb * v;
    f[2] = -0.5f * cc;
    f[3] = cc * v + cb * u;
    f[4] = -cb;
    f[5] = -0.5f * (ca * u * u + cc * v * v) - cb * u * v;
    f[6] = 0.0f;
    f[7] = 0.0f;

    op[n] = 1.0f / (1.0f + __expf(-trans[n]));   // sigmoid

    // real SH basis degree 3, dot with coeffs
    float x = lx, y = ly, z = lz;
    float xx = x * x, yy = y * y, zz = z * z;
    float xy = x * y, yz = y * z, xz = x * z;
    const float* c = sh + n * 16;
    float d0 = 0.28209479177f * c[0]
             + (-0.48860251190f * y) * c[1]
             + ( 0.48860251190f * z) * c[2]
             + (-0.48860251190f * x) * c[3];
    float d1 = ( 1.09254843059f * xy) * c[4]
             + (-1.09254843059f * yz) * c[5]
             + ( 0.31539156525f * (2.0f * zz - xx - yy)) * c[6]
             + (-1.09254843059f * xz) * c[7]
             + ( 0.54627421530f * (xx - yy)) * c[8];
    float d2 = (-0.59004358993f * y * (3.0f * xx - yy)) * c[9]
             + ( 2.89061144264f * xy * z) * c[10]
             + (-0.45704579946f * y * (4.0f * zz - xx - yy)) * c[11]
             + ( 0.37317633259f * z * (2.0f * zz - 3.0f * xx - 3.0f * yy)) * c[12]
             + (-0.45704579946f * x * (4.0f * zz - xx - yy)) * c[13]
             + ( 1.44530572132f * z * (xx - yy)) * c[14]
             + (-0.59004358993f * x * (xx - 3.0f * yy)) * c[15];
    inten[n] = fmaxf(d0 + d1 + d2 + 0.5f, 0.0f);

    rkey[n] = r;
}

// ------------------------------------------------------------------
// Kernel 2: single-block bitonic argsort of 2048 ranges (ascending)
// ------------------------------------------------------------------
__global__ void __launch_bounds__(1024)
sar_sort(const float* __restrict__ rkey, int* __restrict__ order) {
    __shared__ float k[N_GAUSS];
    __shared__ int   id[N_GAUSS];
    int t = threadIdx.x;
    k[t] = rkey[t];               id[t] = t;
    k[t + 1024] = rkey[t + 1024]; id[t + 1024] = t + 1024;
    __syncthreads();
    for (unsigned kk = 2; kk <= N_GAUSS; kk <<= 1) {
        for (unsigned j = kk >> 1; j > 0; j >>= 1) {
            unsigned i = 2u * t - (t & (j - 1));
            unsigned p = i + j;
            bool up = ((i & kk) == 0);
            float ki = k[i], kp = k[p];
            if (up ? (ki > kp) : (ki < kp)) {
                k[i] = kp; k[p] = ki;
                int tmp = id[i]; id[i] = id[p]; id[p] = tmp;
            }
            __syncthreads();
        }
    }
    order[t] = id[t];
    order[t + 1024] = id[t + 1024];
}

// ------------------------------------------------------------------
// Kernel 3: WMMA rasterizer. 8 waves/block, 16 pixels/wave.
// power tile (16 gauss x 16 px) = 2x V_WMMA_F32_16X16X4_F32 (K=8 padded).
// Depth-ordered blend: each lane scans its 8 rows, halves composed via
// shfl_xor(16) (associative (img,T) composition), wave carries running state.
// ------------------------------------------------------------------
__global__ void __launch_bounds__(256)
sar_rasterize(const float* __restrict__ feat,
              const float* __restrict__ op,
              const float* __restrict__ inten,
              const int*   __restrict__ order,
              float* __restrict__ out) {
    __shared__ __align__(16) float s_feat[CHUNK * 8];
    __shared__ float s_op[CHUNK];
    __shared__ float s_in[CHUNK];

    int tid  = threadIdx.x;
    int lane = tid & 31;
    int wave = tid >> 5;
    int half = lane >> 4;      // 0: rows 0-7 (nearer), 1: rows 8-15
    int col  = lane & 15;      // pixel column within tile / gaussian row for A

    int q = (blockIdx.x * 8 + wave) * 16 + col;     // flat pixel id
    float pu = (float)(q >> 7) + 0.5f;              // range axis
    float pv = (float)(q & (AS - 1)) + 0.5f;        // azimuth axis

    // B operand (K x 16 pixel features), K split across lane halves:
    // half0 -> K0,K1 = pu^2, pu ; half1 -> K2,K3 = pv^2, pv
    // half0 -> K4,K5 = pu*pv, 1 ; half1 -> K6,K7 = 0, 0 (padding)
    v2f b0, b1;
    if (half == 0) { b0.x = pu * pu; b0.y = pu; b1.x = pu * pv; b1.y = 1.0f; }
    else           { b0.x = pv * pv; b0.y = pv; b1.x = 0.0f;    b1.y = 0.0f; }

    float T = 1.0f, IMG = 0.0f;

    for (int cb = 0; cb < N_GAUSS; cb += CHUNK) {
        __syncthreads();
        // stage CHUNK gaussians in sorted order into LDS (gather via order[])
        for (int s = tid; s < CHUNK; s += 256) {
            int g = order[cb + s];
            const float4* fg = (const float4*)(feat + g * 8);
            float4 f0 = fg[0], f1 = fg[1];
            float4* d = (float4*)(s_feat + s * 8);
            d[0] = f0; d[1] = f1;
            s_op[s] = op[g];
            s_in[s] = inten[g];
        }
        __syncthreads();

        for (int gt = 0; gt < CHUNK / 16; ++gt) {
            int rowbase = gt * 16;
            // A operand: lane's row = col; halves carry K{0,1}/{2,3} then K{4,5}/{6,7}
            const float* fr = s_feat + (rowbase + col) * 8 + half * 2;
            v2f a0 = *(const v2f*)fr;
            v2f a1 = *(const v2f*)(fr + 4);

            v8f acc = {};
            acc = __builtin_amdgcn_wmma_f32_16x16x4_f32(
                false, a0, false, b0, (short)0, acc, false, false);
            acc = __builtin_amdgcn_wmma_f32_16x16x4_f32(
                false, a1, false, b1, (short)0, acc, false, false);

            // local front-to-back scan over this lane's 8 gaussian rows
            int rb = rowbase + half * 8;
            float Tl = 1.0f, Il = 0.0f;
            #pragma unroll
            for (int i = 0; i < 8; ++i) {
                float pw = fminf(acc[i], 0.0f);
                float g  = __expf(pw);
                float a  = fminf(s_op[rb + i] * g, 0.99f);
                Il += Tl * s_in[rb + i] * a;
                Tl *= (1.0f - a);
            }
            // compose halves in depth order (rows 0-7 before 8-15)
            float Io = __shfl_xor(Il, 16, 32);
            float To = __shfl_xor(Tl, 16, 32);
            float loI = half ? Io : Il;
            float loT = half ? To : Tl;
            float hiI = half ? Il : Io;
            float hiT = half ? Tl : To;
            IMG += T * (loI + loT * hiI);
            T   *= loT * hiT;
        }
    }

    if (half == 0) out[q] = IMG;
}

// ------------------------------------------------------------------
extern "C" void kernel_launch(void* const* d_in, const int* in_sizes, int n_in,
                              void* d_out, int out_size, void* d_ws, size_t ws_size,
                              hipStream_t stream) {
    const float* means = (const float*)d_in[0];
    const float* cov   = (const float*)d_in[1];
    const float* trans = (const float*)d_in[2];
    const float* sh    = (const float*)d_in[3];

    float* ws    = (float*)d_ws;
    float* feat  = ws + WS_FEAT;
    float* opw   = ws + WS_OP;
    float* inten = ws + WS_INT;
    float* rkey  = ws + WS_KEY;
    int*   order = (int*)(ws + WS_ORD);
    float* out   = (float*)d_out;

    hipLaunchKernelGGL(sar_preprocess, dim3(N_GAUSS / 256), dim3(256), 0, stream,
                       means, cov, trans, sh, feat, opw, inten, rkey);
    hipLaunchKernelGGL(sar_sort, dim3(1), dim3(1024), 0, stream, rkey, order);
    hipLaunchKernelGGL(sar_rasterize, dim3(NPIX / 128), dim3(256), 0, stream,
                       feat, opw, inten, order, out);
}